// GaussianRasterizer_26585847562982
// MI455X (gfx1250) — compile-verified
//
#include <hip/hip_runtime.h>
#include <hip/hip_bf16.h>
#include <math.h>

#define NG 1024
#define IH 128
#define IW 128

typedef __attribute__((ext_vector_type(2))) float v2f;
typedef __attribute__((ext_vector_type(8))) float v8f;

// ---------------------------------------------------------------------------
// Kernel 1: per-gaussian preprocess + bitonic depth sort (single block)
// ---------------------------------------------------------------------------
__global__ __launch_bounds__(256) void gs_preprocess_sort(
    const float* __restrict__ means, const float* __restrict__ opac,
    const float* __restrict__ cols,  const float* __restrict__ scales,
    const float* __restrict__ rots,  const float* __restrict__ ev,
    const float* __restrict__ intr,
    float* __restrict__ radii_out,      // d_out + 3*IH*IW  (radii as float)
    float* __restrict__ geo_out,        // d_ws: 5*NG floats (mx,my,g0,g1,g2 SoA, sorted)
    float4* __restrict__ col_out)       // d_ws + 5*NG*4: NG float4 (r,g,b,opac, sorted)
{
  __shared__ float s_key[NG];
  __shared__ int   s_idx[NG];
  __shared__ float s_mx[NG], s_my[NG], s_g0[NG], s_g1[NG], s_g2[NG];
  __shared__ float4 s_c[NG];

  // Camera setup (wave-uniform, cheap to recompute per thread)
  float qx = ev[0], qy = ev[1], qz = ev[2], qw = ev[3];
  float tx = ev[4], ty = ev[5], tzv = ev[6];
  float d2 = qx*qx + qy*qy + qz*qz;
  float m00 = 1.f + 2.f*(qx*qx - d2), m01 = 2.f*(qx*qy - qw*qz), m02 = 2.f*(qx*qz + qw*qy);
  float m10 = 2.f*(qx*qy + qw*qz), m11 = 1.f + 2.f*(qy*qy - d2), m12 = 2.f*(qy*qz - qw*qx);
  float m20 = 2.f*(qx*qz - qw*qy), m21 = 2.f*(qy*qz + qw*qx), m22 = 1.f + 2.f*(qz*qz - d2);
  float tanx = tanf(intr[0] * 0.5f), tany = tanf(intr[4] * 0.5f);
  float focal_x = (float)IW / (2.f * tanx), focal_y = (float)IH / (2.f * tany);
  float itx = 1.f / tanx, ity = 1.f / tany;
  float limx = 1.3f * tanx, limy = 1.3f * tany;

  for (int i = threadIdx.x; i < NG; i += 256) {
    float X = means[3*i], Y = means[3*i+1], Z = means[3*i+2];
    float pcx = m00*X + m01*Y + m02*Z + tx;
    float pcy = m10*X + m11*Y + m12*Z + ty;
    float pcz = m20*X + m21*Y + m22*Z + tzv;
    float depth = pcz;
    // full = P @ M :  row0 = itx*Mrow0, row1 = ity*Mrow1, row3 = Mrow2
    float ph0 = itx * pcx;
    float ph1 = ity * pcy;
    float pw  = 1.f / (pcz + 1e-7f);
    float mx = ((ph0*pw + 1.f) * (float)IW - 1.f) * 0.5f;
    float my = ((ph1*pw + 1.f) * (float)IH - 1.f) * 0.5f;

    float r0 = rots[4*i], r1 = rots[4*i+1], r2 = rots[4*i+2], r3 = rots[4*i+3];
    float invn = 1.f / sqrtf(r0*r0 + r1*r1 + r2*r2 + r3*r3);
    float qr = r0*invn, ax = r1*invn, ay = r2*invn, az = r3*invn;
    float R00 = 1.f-2.f*(ay*ay+az*az), R01 = 2.f*(ax*ay-qr*az), R02 = 2.f*(ax*az+qr*ay);
    float R10 = 2.f*(ax*ay+qr*az), R11 = 1.f-2.f*(ax*ax+az*az), R12 = 2.f*(ay*az-qr*ax);
    float R20 = 2.f*(ax*az-qr*ay), R21 = 2.f*(ay*az+qr*ax), R22 = 1.f-2.f*(ax*ax+ay*ay);
    float sa = scales[3*i], sb = scales[3*i+1], sc = scales[3*i+2];
    float w0 = sa*sa, w1 = sb*sb, w2 = sc*sc;
    float S00 = R00*R00*w0 + R01*R01*w1 + R02*R02*w2;
    float S01 = R00*R10*w0 + R01*R11*w1 + R02*R12*w2;
    float S02 = R00*R20*w0 + R01*R21*w1 + R02*R22*w2;
    float S11 = R10*R10*w0 + R11*R11*w1 + R12*R12*w2;
    float S12 = R10*R20*w0 + R11*R21*w1 + R12*R22*w2;
    float S22 = R20*R20*w0 + R21*R21*w1 + R22*R22*w2;

    float tzi = 1.f / depth;
    float txc = fminf(fmaxf(pcx*tzi, -limx), limx) * depth;
    float tyc = fminf(fmaxf(pcy*tzi, -limy), limy) * depth;
    float J00 = focal_x*tzi, J02 = -focal_x*txc*(tzi*tzi);
    float J11 = focal_y*tzi, J12 = -focal_y*tyc*(tzi*tzi);
    // T3 = J @ Rcw
    float T00 = J00*m00 + J02*m20, T01 = J00*m01 + J02*m21, T02 = J00*m02 + J02*m22;
    float T10 = J11*m10 + J12*m20, T11 = J11*m11 + J12*m21, T12 = J11*m12 + J12*m22;
    // cov2 = T3 Sigma T3^T
    float A0 = T00*S00 + T01*S01 + T02*S02;
    float A1 = T00*S01 + T01*S11 + T02*S12;
    float A2 = T00*S02 + T01*S12 + T02*S22;
    float B0 = T10*S00 + T11*S01 + T12*S02;
    float B1 = T10*S01 + T11*S11 + T12*S12;
    float B2 = T10*S02 + T11*S12 + T12*S22;
    float c00 = A0*T00 + A1*T01 + A2*T02 + 0.3f;
    float c01 = A0*T10 + A1*T11 + A2*T12;
    float c11 = B0*T10 + B1*T11 + B2*T12 + 0.3f;
    float det = c00*c11 - c01*c01;
    bool valid = (depth > 0.2f) && (det > 0.0f);
    float inv_det = (det != 0.0f) ? (1.f / det) : 0.f;
    float conA = c11*inv_det, conB = -c01*inv_det, conC = c00*inv_det;
    float mid = 0.5f*(c00 + c11);
    float lam1 = mid + sqrtf(fmaxf(0.1f, mid*mid - det));
    int rad = valid ? (int)ceilf(3.f * sqrtf(lam1)) : 0;
    radii_out[i] = (float)rad;

    s_key[i] = depth; s_idx[i] = i;
    s_mx[i] = mx; s_my[i] = my;
    s_g0[i] = -0.5f*conA; s_g1[i] = -0.5f*conC; s_g2[i] = -conB;
    s_c[i] = make_float4(cols[3*i], cols[3*i+1], cols[3*i+2], valid ? opac[i] : 0.f);
  }
  __syncthreads();

  // Bitonic sort ascending by depth (1024 = 2^10, disjoint pairs per pass)
  for (int k = 2; k <= NG; k <<= 1) {
    for (int j = k >> 1; j > 0; j >>= 1) {
      for (int i = threadIdx.x; i < NG; i += 256) {
        int p = i ^ j;
        if (p > i) {
          bool asc = ((i & k) == 0);
          float ka = s_key[i], kb = s_key[p];
          if (asc ? (ka > kb) : (ka < kb)) {
            s_key[i] = kb; s_key[p] = ka;
            int t = s_idx[i]; s_idx[i] = s_idx[p]; s_idx[p] = t;
          }
        }
      }
      __syncthreads();
    }
  }
  for (int i = threadIdx.x; i < NG; i += 256) {
    int j = s_idx[i];
    geo_out[i]        = s_mx[j];
    geo_out[NG + i]   = s_my[j];
    geo_out[2*NG + i] = s_g0[j];
    geo_out[3*NG + i] = s_g1[j];
    geo_out[4*NG + i] = s_g2[j];
    col_out[i]        = s_c[j];
  }
}

// ---------------------------------------------------------------------------
// Kernel 2: rasterize. One wave = one 32-pixel row strip. Power via WMMA f32.
// ---------------------------------------------------------------------------
__global__ __launch_bounds__(256) void gs_raster(
    const float* __restrict__ geo, const float4* __restrict__ colop,
    const float* __restrict__ bg, float* __restrict__ out)
{
  __shared__ float  sg[5*NG];   // mx,my,g0,g1,g2 SoA (sorted)
  __shared__ float4 sc[NG];     // r,g,b,opac       (sorted)
  for (int i = threadIdx.x; i < 5*NG; i += 256) sg[i] = geo[i];
  for (int i = threadIdx.x; i < NG; i += 256)   sc[i] = colop[i];
  __syncthreads();

  int wave = (blockIdx.x * 256 + (int)threadIdx.x) >> 5;   // 0..511
  int lane = threadIdx.x & 31;
  int y  = wave >> 2;
  int x0 = (wave & 3) << 5;
  bool lo = lane < 16;
  float cx0 = (float)x0 + 15.5f;
  float fyy = (float)y;

  // B operands (4xK rows striped: vgpr r = rows K=r (lanes 0-15) / K=r+2 (16-31))
  float xl  = (float)(lane & 15);
  float px1 = xl - 15.5f;   // tile1 pixel offsets
  float px2 = xl + 0.5f;    // tile2 pixel offsets
  v2f b1, b2;
  b1[0] = lo ? px1*px1 : 1.0f;  b1[1] = lo ? px1 : 0.0f;
  b2[0] = lo ? px2*px2 : 1.0f;  b2[1] = lo ? px2 : 0.0f;

  float T = 1.0f, cr = 0.f, cg = 0.f, cb = 0.f;
  bool alive = true;

  for (int base = 0; base < NG; base += 16) {
    int g = base + (lane & 15);
    float mxp = sg[g]      - cx0;
    float myp = sg[NG + g] - fyy;
    float G0 = sg[2*NG + g];
    float G1 = sg[3*NG + g];
    float G2 = sg[4*NG + g];
    // power(px') = G0*px'^2 + g3*px' + g5   (py' == 0 on this row)
    float g3 = -2.f*G0*mxp - G2*myp;
    float g5 = (G0*mxp)*mxp + (G1*myp)*myp + (G2*mxp)*myp;
    // A operand: 16x4 f32; lanes 0-15 hold K0,K1; lanes 16-31 hold K2,K3
    v2f a;
    a[0] = lo ? G0 : g5;
    a[1] = lo ? g3 : 0.f;

    v8f zero = {};
    v8f DA = __builtin_amdgcn_wmma_f32_16x16x4_f32(false, a, false, b1,
                                                   (short)0, zero, false, false);
    v8f DB = __builtin_amdgcn_wmma_f32_16x16x4_f32(false, a, false, b2,
                                                   (short)0, zero, false, false);

    // Redistribute so lane l owns pixel x0+l with gaussians base..base+15 in order.
    float P[16];
#pragma unroll
    for (int r = 0; r < 8; ++r) {
      float da = DA[r], db = DB[r];
      float das = __int_as_float(__builtin_amdgcn_ds_swizzle(__float_as_int(da), 0x401f)); // SWAPX16
      float dbs = __int_as_float(__builtin_amdgcn_ds_swizzle(__float_as_int(db), 0x401f));
      P[r]     = lo ? da  : dbs;
      P[r + 8] = lo ? das : db;
    }

    // Sequential front-to-back composite (exact reference semantics).
    // __expf -> native v_exp_f32; its ~1e-6 rel error is negligible vs the
    // quadratic-expansion rounding already present, and edge cases
    // (exp(-inf)=0, NaN gated by p<=0, inf clamped then killed by ok) match.
#pragma unroll
    for (int r = 0; r < 16; ++r) {
      float4 c4 = sc[base + r];
      float p = P[r];
      float alpha = fminf(0.99f, c4.w * __expf(p));
      bool ok = (p <= 0.0f) && (alpha >= (1.0f / 255.0f));
      alpha = ok ? alpha : 0.0f;
      if (alive && alpha > 0.0f) {
        float test = T * (1.0f - alpha);
        if (test < 1e-4f) {
          alive = false;                 // gaussian not applied, all later dead
        } else {
          float w = T * alpha;
          cr += w * c4.x; cg += w * c4.y; cb += w * c4.z;
          T = test;
        }
      }
    }
    if (__ballot(alive) == 0) break;     // wave-uniform: EXEC stays full for WMMA
  }

  int pix = y * IW + x0 + lane;
  out[pix]             = cr + T * bg[0];
  out[IH*IW + pix]     = cg + T * bg[1];
  out[2*IH*IW + pix]   = cb + T * bg[2];
}

// ---------------------------------------------------------------------------
extern "C" void kernel_launch(void* const* d_in, const int* in_sizes, int n_in,
                              void* d_out, int out_size, void* d_ws, size_t ws_size,
                              hipStream_t stream) {
  const float* means  = (const float*)d_in[0];
  const float* opac   = (const float*)d_in[1];
  const float* cols   = (const float*)d_in[2];
  const float* scales = (const float*)d_in[3];
  const float* rots   = (const float*)d_in[4];
  const float* ev     = (const float*)d_in[5];
  const float* bg     = (const float*)d_in[6];
  const float* intr   = (const float*)d_in[7];

  float*  out   = (float*)d_out;
  float*  radii = out + 3 * IH * IW;                 // radii as float values
  float*  geo   = (float*)d_ws;                      // 5*NG floats
  float4* colop = (float4*)((char*)d_ws + 5 * NG * sizeof(float)); // NG float4

  gs_preprocess_sort<<<1, 256, 0, stream>>>(means, opac, cols, scales, rots, ev,
                                            intr, radii, geo, colop);
  gs_raster<<<64, 256, 0, stream>>>(geo, colop, bg, out);
}